// IVDetect_43482248905417
// MI455X (gfx1250) — compile-verified
//
#include <hip/hip_runtime.h>

#define N_NODES  50000
#define N_EDGES  800000
#define IN_DIM   128
#define HID      64
#define N_CLS    2
#define N_GRAPHS 256

typedef __attribute__((ext_vector_type(2))) float v2f;
typedef __attribute__((ext_vector_type(8))) float v8f;

// ---------------------------------------------------------------------------
// float atomic max via integer punning (works for mixed signs, init = -inf)
// ---------------------------------------------------------------------------
__device__ __forceinline__ void atomicMaxFloat(float* addr, float val) {
    if (val >= 0.0f)
        atomicMax((int*)addr, __float_as_int(val));
    else
        atomicMin((unsigned int*)addr, __float_as_uint(val));
}

// ---------------------------------------------------------------------------
// small utility kernels
// ---------------------------------------------------------------------------
__global__ void k_zero(float* __restrict__ p, int n) {
    int i = blockIdx.x * blockDim.x + threadIdx.x;
    if (i < n) p[i] = 0.0f;
}

__global__ void k_init_neg_inf(float* __restrict__ p, int n) {
    int i = blockIdx.x * blockDim.x + threadIdx.x;
    if (i < n) p[i] = -__builtin_inff();
}

// deg[dst] += 1 per edge
__global__ void k_deg(const int* __restrict__ dst, float* __restrict__ deg, int ne) {
    int e = blockIdx.x * blockDim.x + threadIdx.x;
    if (e < ne) atomicAdd(&deg[dst[e]], 1.0f);
}

// dis = rsqrt(deg + 1)
__global__ void k_dis(float* __restrict__ deg_dis, int n) {
    int i = blockIdx.x * blockDim.x + threadIdx.x;
    if (i < n) deg_dis[i] = rsqrtf(deg_dis[i] + 1.0f);
}

// ---------------------------------------------------------------------------
// fp32 WMMA GEMM, N fixed = 64 (HID). One wave computes a full 16(M) x 64(N)
// output strip: the A fragment is loaded ONCE per k-step and reused by 4
// V_WMMA_F32_16X16X4_F32 issues (4 independent accumulator chains).
//   A: [M x K] row-major (node features), B: [K x 64] row-major (weights)
// Epilogue:
//   AGG != nullptr (GCN conv pre-pass):  T = A@B ;  AGG = (A@B) * dis[m]^2
//   AGG == nullptr (plain linear):       T = A@B + bias[n]
// A-matrix 16x4 layout:  lane = {kh=lane[4], m=lane[3:0]}, regs = K (k0+2kh, +1)
// B-matrix 4x16 layout:  lane = {kh, n},                  regs = K (k0+2kh, +1)
// C 16x16 layout: reg r -> row (r + 8*kh), col = n
// ---------------------------------------------------------------------------
__global__ __launch_bounds__(32)
void k_gemm_wmma(const float* __restrict__ A, const float* __restrict__ B,
                 float* __restrict__ T, float* __restrict__ AGG,
                 const float* __restrict__ dis, const float* __restrict__ bias,
                 int K) {
    const int lane  = threadIdx.x;
    const int m16   = lane & 15;
    const int kh    = lane >> 4;
    const int mbase = blockIdx.x * 16;

    v8f acc0 = {0.f, 0.f, 0.f, 0.f, 0.f, 0.f, 0.f, 0.f};
    v8f acc1 = acc0, acc2 = acc0, acc3 = acc0;

    const float* arow = A + (size_t)(mbase + m16) * K;

    for (int k0 = 0; k0 < K; k0 += 4) {
        const int ka = k0 + 2 * kh;
        v2f a;
        a.x = arow[ka];
        a.y = arow[ka + 1];

        const float* br0 = B + (size_t)ka * HID + m16;        // row ka
        const float* br1 = B + (size_t)(ka + 1) * HID + m16;  // row ka+1
        v2f b0, b1, b2, b3;
        b0.x = br0[0];   b0.y = br1[0];
        b1.x = br0[16];  b1.y = br1[16];
        b2.x = br0[32];  b2.y = br1[32];
        b3.x = br0[48];  b3.y = br1[48];

        acc0 = __builtin_amdgcn_wmma_f32_16x16x4_f32(false, a, false, b0, (short)0, acc0, false, false);
        acc1 = __builtin_amdgcn_wmma_f32_16x16x4_f32(false, a, false, b1, (short)0, acc1, false, false);
        acc2 = __builtin_amdgcn_wmma_f32_16x16x4_f32(false, a, false, b2, (short)0, acc2, false, false);
        acc3 = __builtin_amdgcn_wmma_f32_16x16x4_f32(false, a, false, b3, (short)0, acc3, false, false);
    }

    // epilogue: 4 N-tiles, 8 rows each
    if (AGG != nullptr) {
#pragma unroll
        for (int r = 0; r < 8; ++r) {
            const int m = mbase + 8 * kh + r;
            const float d  = dis[m];
            const float dd = d * d;
            float* trow = T   + (size_t)m * HID + m16;
            float* grow = AGG + (size_t)m * HID + m16;
            const float v0 = acc0[r], v1 = acc1[r], v2 = acc2[r], v3 = acc3[r];
            trow[0]  = v0;       trow[16] = v1;       trow[32] = v2;       trow[48] = v3;
            grow[0]  = v0 * dd;  grow[16] = v1 * dd;  grow[32] = v2 * dd;  grow[48] = v3 * dd;
        }
    } else {
        const float bi0 = bias[m16];
        const float bi1 = bias[m16 + 16];
        const float bi2 = bias[m16 + 32];
        const float bi3 = bias[m16 + 48];
#pragma unroll
        for (int r = 0; r < 8; ++r) {
            const int m = mbase + 8 * kh + r;
            float* trow = T + (size_t)m * HID + m16;
            trow[0]  = acc0[r] + bi0;
            trow[16] = acc1[r] + bi1;
            trow[32] = acc2[r] + bi2;
            trow[48] = acc3[r] + bi3;
        }
    }
}

// ---------------------------------------------------------------------------
// edge scatter for F=64: 16 threads/edge, 4 features each (float4 gather)
//   AGG[dst] += T[src] * dis[src]*dis[dst]
// ---------------------------------------------------------------------------
__global__ void k_edge_scatter64(const int* __restrict__ src, const int* __restrict__ dst,
                                 const float* __restrict__ dis, const float* __restrict__ T,
                                 float* __restrict__ AGG) {
    const int idx = blockIdx.x * blockDim.x + threadIdx.x;
    const int e = idx >> 4;
    if (e >= N_EDGES) return;
    const int f = (idx & 15) << 2;
    const int s = src[e];
    const int d = dst[e];
    const float c = dis[s] * dis[d];
    const float4 v = *(const float4*)(T + (size_t)s * HID + f);
    float* o = AGG + (size_t)d * HID + f;
    atomicAdd(o + 0, v.x * c);
    atomicAdd(o + 1, v.y * c);
    atomicAdd(o + 2, v.z * c);
    atomicAdd(o + 3, v.w * c);
}

// h = relu(AGG + bias), in place
__global__ void k_bias_relu(float* __restrict__ AGG, const float* __restrict__ bias) {
    const int idx = blockIdx.x * blockDim.x + threadIdx.x;
    if (idx >= N_NODES * HID) return;
    const float v = AGG[idx] + bias[idx & (HID - 1)];
    AGG[idx] = v > 0.0f ? v : 0.0f;
}

// ---------------------------------------------------------------------------
// layer 3 projection (HID -> 2): per-node dual dot product + self-loop init
// ---------------------------------------------------------------------------
__global__ void k_lin3(const float* __restrict__ H, const float* __restrict__ W3,
                       const float* __restrict__ dis,
                       float* __restrict__ T3, float* __restrict__ A3) {
    const int m = blockIdx.x * blockDim.x + threadIdx.x;
    if (m >= N_NODES) return;
    const float* h = H + (size_t)m * HID;
    float a0 = 0.0f, a1 = 0.0f;
#pragma unroll 8
    for (int k = 0; k < HID; ++k) {
        const float hv = h[k];
        a0 += hv * W3[k * 2 + 0];
        a1 += hv * W3[k * 2 + 1];
    }
    const float d  = dis[m];
    const float dd = d * d;
    T3[m * 2 + 0] = a0;
    T3[m * 2 + 1] = a1;
    A3[m * 2 + 0] = a0 * dd;
    A3[m * 2 + 1] = a1 * dd;
}

__global__ void k_edge_scatter2(const int* __restrict__ src, const int* __restrict__ dst,
                                const float* __restrict__ dis, const float* __restrict__ T3,
                                float* __restrict__ A3) {
    const int e = blockIdx.x * blockDim.x + threadIdx.x;
    if (e >= N_EDGES) return;
    const int s = src[e];
    const int d = dst[e];
    const float c = dis[s] * dis[d];
    atomicAdd(&A3[d * 2 + 0], T3[s * 2 + 0] * c);
    atomicAdd(&A3[d * 2 + 1], T3[s * 2 + 1] * c);
}

// segment_max pooling: out[batch[m]][c] = max(A3[m][c] + b3[c])
__global__ void k_pool(const float* __restrict__ A3, const float* __restrict__ b3,
                       const int* __restrict__ batch, float* __restrict__ out) {
    const int idx = blockIdx.x * blockDim.x + threadIdx.x;
    if (idx >= N_NODES * N_CLS) return;
    const int m = idx >> 1;
    const int c = idx & 1;
    atomicMaxFloat(&out[batch[m] * N_CLS + c], A3[idx] + b3[c]);
}

// empty-graph guard: non-finite -> 0
__global__ void k_pool_fix(float* __restrict__ out) {
    const int i = blockIdx.x * blockDim.x + threadIdx.x;
    if (i < N_GRAPHS * N_CLS) {
        const float v = out[i];
        if (!(v > -__builtin_inff() && v < __builtin_inff())) out[i] = 0.0f;
    }
}

// ---------------------------------------------------------------------------
// launch
// ---------------------------------------------------------------------------
extern "C" void kernel_launch(void* const* d_in, const int* in_sizes, int n_in,
                              void* d_out, int out_size, void* d_ws, size_t ws_size,
                              hipStream_t stream) {
    (void)in_sizes; (void)n_in; (void)out_size; (void)ws_size;

    const float* x   = (const float*)d_in[0];
    const int*   ei  = (const int*)d_in[1];      // [2, N_EDGES]: row0 = src, row1 = dst
    const int*   bat = (const int*)d_in[2];
    const float* W1  = (const float*)d_in[3];
    const float* b1  = (const float*)d_in[4];
    const float* Wc  = (const float*)d_in[5];
    const float* bc  = (const float*)d_in[6];
    const float* W2  = (const float*)d_in[7];
    const float* b2  = (const float*)d_in[8];
    const float* W3  = (const float*)d_in[9];
    const float* b3  = (const float*)d_in[10];
    float* out = (float*)d_out;

    const int* src = ei;
    const int* dst = ei + N_EDGES;

    // workspace layout (floats), 16B-aligned blocks
    float* ws  = (float*)d_ws;
    float* dis = ws;                              // 50048 (deg, then rsqrt in place)
    float* T   = dis + 50048;                     // 3,200,000  projected features
    float* AG  = T   + (size_t)N_NODES * HID;     // 3,200,000  aggregation / h buffer
    float* HB  = AG  + (size_t)N_NODES * HID;     // 3,200,000  center-linear output
    float* T3  = HB  + (size_t)N_NODES * HID;     // 100,000
    float* A3  = T3  + (size_t)N_NODES * N_CLS;   // 100,000

    const int BT = 256;
    const int gNode  = (N_NODES + BT - 1) / BT;
    const int gEdge  = (N_EDGES + BT - 1) / BT;
    const int gElem  = (N_NODES * HID + BT - 1) / BT;
    const int gScat  = (N_EDGES * 16 + BT - 1) / BT;
    const int gPool  = (N_NODES * N_CLS + BT - 1) / BT;
    const int gGemm  = N_NODES / 16;              // 3125 waves, one 16x64 strip each

    // degree -> dis ; output init
    k_zero<<<gNode, BT, 0, stream>>>(dis, N_NODES);
    k_init_neg_inf<<<2, BT, 0, stream>>>(out, N_GRAPHS * N_CLS);
    k_deg<<<gEdge, BT, 0, stream>>>(dst, dis, N_EDGES);
    k_dis<<<gNode, BT, 0, stream>>>(dis, N_NODES);

    // layer 1: t = x@W1 ; agg = t*dis^2 + scatter ; h1 = relu(agg + b1) (in AG)
    k_gemm_wmma<<<gGemm, 32, 0, stream>>>(x, W1, T, AG, dis, nullptr, IN_DIM);
    k_edge_scatter64<<<gScat, BT, 0, stream>>>(src, dst, dis, T, AG);
    k_bias_relu<<<gElem, BT, 0, stream>>>(AG, b1);

    // center linear: HB = h1@Wc + bc
    k_gemm_wmma<<<gGemm, 32, 0, stream>>>(AG, Wc, HB, nullptr, nullptr, bc, HID);

    // layer 2: t = HB@W2 ; agg (reuse AG) ; h2 = relu(agg + b2) (in AG)
    k_gemm_wmma<<<gGemm, 32, 0, stream>>>(HB, W2, T, AG, dis, nullptr, HID);
    k_edge_scatter64<<<gScat, BT, 0, stream>>>(src, dst, dis, T, AG);
    k_bias_relu<<<gElem, BT, 0, stream>>>(AG, b2);

    // layer 3: t3 = h2@W3 ; a3 = t3*dis^2 + scatter ; pool max(a3 + b3)
    k_lin3<<<gNode, BT, 0, stream>>>(AG, W3, dis, T3, A3);
    k_edge_scatter2<<<gEdge, BT, 0, stream>>>(src, dst, dis, T3, A3);
    k_pool<<<gPool, BT, 0, stream>>>(A3, b3, bat, out);
    k_pool_fix<<<2, BT, 0, stream>>>(out);
}